// DLTPoseSolver_13494787244423
// MI455X (gfx1250) — compile-verified
//
#include <hip/hip_runtime.h>

#define B_BATCH 512
#define N_PTS   1024

typedef __attribute__((ext_vector_type(2))) float v2f;
typedef __attribute__((ext_vector_type(8))) float v8f;

// Kernel 1: per-batch Gram matrix M = A^T A (16x16, 12x12 active) via fp32 WMMA.
// Block = 256 threads (8 waves), one block per batch. Each wave handles 128 points.
//
// Vandermonde rows for point (x,y,z,u,v), weight w:
//   type 0: w*[0,0,0,0, -x,-y,-z,-1,  vx,vy,vz,v ]
//   type 1: w*[x,y,z,1,  0,0,0,0,  -ux,-uy,-uz,-u]
//   type 2: w*[-vx,-vy,-vz,-v, ux,uy,uz,u, 0,0,0,0]
// Every element factors as (w*base) * (sc + su*u + sv*v) with base in {x,y,z,1}
// and sc,su,sv in {0,+-1} -- all loop-invariant per lane. LDS records store
// {w*x, w*y, w*z, w, u, v} so the inner loop is branchless.
__global__ __launch_bounds__(256)
void gram_kernel(const float* __restrict__ weights,
                 const float* __restrict__ pts,
                 float* __restrict__ gram_out) {
    __shared__ float lpts[N_PTS * 8];   // 32 KB: {wx,wy,wz,w,u,v,0,0} per point
    __shared__ float lgram[256];

    const int b = blockIdx.x;
    const int t = threadIdx.x;

    // Stage + pre-scale points into LDS.
    for (int i = t; i < N_PTS; i += 256) {
        const float* p = pts + ((size_t)b * N_PTS + i) * 5;
        const float x = p[0], y = p[1], z = p[2], u = p[3], v = p[4];
        const float w = weights[(size_t)b * N_PTS + i];
        float* rec = &lpts[i * 8];
        rec[0] = w * x; rec[1] = w * y; rec[2] = w * z; rec[3] = w;
        rec[4] = u;     rec[5] = v;     rec[6] = 0.0f;  rec[7] = 0.0f;
    }
    lgram[t] = 0.0f;
    __syncthreads();

    const int wave = t >> 5;      // 0..7 (wave32)
    const int lane = t & 31;
    const int m    = lane & 15;   // M (and N) index 0..15
    const int half = lane >> 4;   // selects K rows {0,1} vs {2,3}
    const int mi   = m & 3;       // which of {wx,wy,wz,w} this lane reads
    const int blk  = m >> 2;      // 0..3 (3 => padding lane, coeffs all zero)

    // Per-lane loop-invariant tables for the 3 chunks x 2 VGPR slots.
    // A-matrix layout (16x4 f32): VGPR0 holds K rows {0,2} by lane half,
    // VGPR1 holds K rows {1,3}. Symmetric product => same image for A and B.
    int   dp[3][2];
    float scv[3][2], suv[3][2], svv[3][2];
    #pragma unroll
    for (int cc = 0; cc < 3; ++cc) {
        #pragma unroll
        for (int slot = 0; slot < 2; ++slot) {
            const int rr = cc * 4 + (half ? 2 : 0) + slot;  // row within 12-row group
            dp[cc][slot] = rr / 3;                          // point offset 0..3
            const int ty = rr % 3;
            float sc = 0.0f, su = 0.0f, sv = 0.0f;
            if (blk < 3) {  // m < 12
                if (ty == 0)      { if (blk == 1) sc = -1.0f; else if (blk == 2) sv =  1.0f; }
                else if (ty == 1) { if (blk == 0) sc =  1.0f; else if (blk == 2) su = -1.0f; }
                else              { if (blk == 0) sv = -1.0f; else if (blk == 1) su =  1.0f; }
            }
            scv[cc][slot] = sc; suv[cc][slot] = su; svv[cc][slot] = sv;
        }
    }

    v8f acc = {};
    const int pbase = wave * 128;
    for (int g = 0; g < 32; ++g) {
        const int gp = pbase + g * 4;   // 4 points -> 12 rows -> 3 K=4 chunks
        #pragma unroll
        for (int cc = 0; cc < 3; ++cc) {
            v2f a;
            #pragma unroll
            for (int slot = 0; slot < 2; ++slot) {
                const float* rec = &lpts[(gp + dp[cc][slot]) * 8];
                const float basew = rec[mi];
                const float uu    = rec[4];
                const float vv    = rec[5];
                const float coef  = fmaf(suv[cc][slot], uu,
                                    fmaf(svv[cc][slot], vv, scv[cc][slot]));
                const float e = basew * coef;
                if (slot == 0) a.x = e; else a.y = e;
            }
            acc = __builtin_amdgcn_wmma_f32_16x16x4_f32(
                      /*neg_a=*/false, a, /*neg_b=*/false, a,
                      /*c_mod=*/(short)0, acc, /*reuse_a=*/false, /*reuse_b=*/false);
        }
    }

    // Cross-wave reduction of the 16x16 accumulator tiles.
    // C/D layout: VGPR v, lane l -> D[M = v + 8*(l>>4)][N = l&15].
    #pragma unroll
    for (int vv = 0; vv < 8; ++vv) {
        const int Mrow = vv + 8 * half;
        atomicAdd(&lgram[Mrow * 16 + m], acc[vv]);
    }
    __syncthreads();
    gram_out[(size_t)b * 256 + t] = lgram[t];
}

// Kernel 2: one wave per batch. Jacobi eigensolver on the 12x12 Gram,
// then emit pose_raw, pose_norm = pose_raw @ norm_mat, and sorted singular values.
__global__ __launch_bounds__(32)
void jacobi_kernel(const float* __restrict__ gram,
                   const float* __restrict__ norm_mat,
                   float* __restrict__ out) {
    __shared__ float A[16 * 16];
    __shared__ float V[12 * 13];
    __shared__ float pr[12];
    __shared__ float sv[12];

    const int b = blockIdx.x;
    const int i = threadIdx.x;

    for (int k = i; k < 256; k += 32) A[k] = gram[(size_t)b * 256 + k];
    if (i < 12)
        for (int j = 0; j < 12; ++j) V[i * 13 + j] = (i == j) ? 1.0f : 0.0f;
    __syncthreads();

    for (int sweep = 0; sweep < 10; ++sweep) {
        for (int p = 0; p < 11; ++p) {
            for (int q = p + 1; q < 12; ++q) {
                // All lanes compute the same rotation from LDS (no broadcast needed).
                const float app = A[p * 16 + p];
                const float aqq = A[q * 16 + q];
                const float apq = A[p * 16 + q];
                float c = 1.0f, s = 0.0f;
                if (fabsf(apq) > 1e-20f) {
                    const float tau = (aqq - app) / (2.0f * apq);
                    const float tt  = (tau >= 0.0f ? 1.0f : -1.0f) /
                                      (fabsf(tau) + sqrtf(1.0f + tau * tau));
                    c = rsqrtf(1.0f + tt * tt);
                    s = tt * c;
                }
                __syncthreads();
                if (i < 12) {                       // B = A * J (columns p,q)
                    const float aip = A[i * 16 + p];
                    const float aiq = A[i * 16 + q];
                    A[i * 16 + p] = c * aip - s * aiq;
                    A[i * 16 + q] = s * aip + c * aiq;
                }
                __syncthreads();
                if (i < 12) {                       // A' = J^T * B (rows p,q)
                    const float api = A[p * 16 + i];
                    const float aqi = A[q * 16 + i];
                    A[p * 16 + i] = c * api - s * aqi;
                    A[q * 16 + i] = s * api + c * aqi;
                }
                __syncthreads();
                if (i < 12) {                       // accumulate eigenvectors
                    const float vip = V[i * 13 + p];
                    const float viq = V[i * 13 + q];
                    V[i * 13 + p] = c * vip - s * viq;
                    V[i * 13 + q] = s * vip + c * viq;
                }
                __syncthreads();
            }
        }
    }

    if (i < 12) sv[i] = sqrtf(fmaxf(A[i * 16 + i], 0.0f));
    __syncthreads();

    // Eigenvector of the smallest singular value.
    int jmin = 0;
    float smin = sv[0];
    for (int j = 1; j < 12; ++j) if (sv[j] < smin) { smin = sv[j]; jmin = j; }
    if (i < 12) pr[i] = V[i * 13 + jmin];
    __syncthreads();

    // Deterministic sign convention: largest-magnitude element positive.
    int kmax = 0; float amax = fabsf(pr[0]);
    for (int j = 1; j < 12; ++j) {
        const float aj = fabsf(pr[j]);
        if (aj > amax) { amax = aj; kmax = j; }
    }
    const float sgn = (pr[kmax] < 0.0f) ? -1.0f : 1.0f;
    __syncthreads();
    if (i < 12) pr[i] *= sgn;
    __syncthreads();

    if (i < 12) {
        // pose_raw  [B,3,4] at offset 0
        out[(size_t)b * 12 + i] = pr[i];
        // pose_norm [B,3,4] at offset B*12 : (3x4) @ (4x4)
        const int r = i >> 2, cc = i & 3;
        const float* nm = norm_mat + (size_t)b * 16;
        float accv = 0.0f;
        #pragma unroll
        for (int k = 0; k < 4; ++k) accv += pr[r * 4 + k] * nm[k * 4 + cc];
        out[(size_t)B_BATCH * 12 + (size_t)b * 12 + i] = accv;
        // s [B,12] at offset B*24, sorted descending via rank counting.
        const float si = sv[i];
        int rank = 0;
        for (int j = 0; j < 12; ++j) {
            const float sj = sv[j];
            if (sj > si || (sj == si && j < i)) ++rank;
        }
        out[(size_t)B_BATCH * 24 + (size_t)b * 12 + rank] = si;
    }
}

extern "C" void kernel_launch(void* const* d_in, const int* in_sizes, int n_in,
                              void* d_out, int out_size, void* d_ws, size_t ws_size,
                              hipStream_t stream) {
    (void)in_sizes; (void)n_in; (void)out_size; (void)ws_size;
    const float* weights = (const float*)d_in[0];   // [B,N]
    const float* pts     = (const float*)d_in[1];   // [B,N,5]
    const float* normm   = (const float*)d_in[2];   // [B,4,4]
    float* out  = (float*)d_out;                    // pose_raw | pose_norm | s
    float* gram = (float*)d_ws;                     // B * 256 floats = 512 KB

    gram_kernel<<<B_BATCH, 256, 0, stream>>>(weights, pts, gram);
    jacobi_kernel<<<B_BATCH, 32, 0, stream>>>(gram, normm, out);
}